// GAElayer_5592047419801
// MI455X (gfx1250) — compile-verified
//
#include <hip/hip_runtime.h>
#include <math.h>

// Problem constants (from reference)
#define N_NODES 50000
#define DIN     128
#define DOUT    64
#define KHALF   5
#define WINN    10      // 2K neighbors per node
#define TILE    16      // nodes per block == one WMMA M-tile
#define WROWS   26      // max window rows per block (15 + 2K + 1)
#define WS      132     // LDS stride (floats) for window rows (pad: bank spread + 16B align)
#define AS      132     // LDS stride for agg tile
#define ES      68      // LDS stride for enc tile

typedef float v2f __attribute__((ext_vector_type(2)));
typedef float v8f __attribute__((ext_vector_type(8)));
typedef int   v4i __attribute__((vector_size(16)));   // matches async builtin pointee
typedef __attribute__((address_space(1))) v4i* gv4i_p;  // global (prints as __device__)
typedef __attribute__((address_space(3))) v4i* lv4i_p;  // LDS (prints as __shared__)

static __device__ __forceinline__ v8f wmma_f32(v2f a, v2f b, v8f c) {
  // D = A(16x4 f32) * B(4x16 f32) + C(16x16 f32)
  return __builtin_amdgcn_wmma_f32_16x16x4_f32(false, a, false, b, (short)0, c, false, false);
}

__global__ __launch_bounds__(128) void gae_fused_kernel(
    const float* __restrict__ h,  const float* __restrict__ We,
    const float* __restrict__ be, const float* __restrict__ Wd,
    const float* __restrict__ bd, float* __restrict__ out)
{
  __shared__ float winb[WROWS * WS];  // h window tile; reused as dec staging tile
  __shared__ float aggb[TILE * AS];   // aggregated features [16][128] (padded)
  __shared__ float encb[TILE * ES];   // encoder output [16][64] (padded)

  const int tid = threadIdx.x;
  const int s   = blockIdx.x * TILE;  // first node of this block

  const int wave = tid >> 5;
  const int lane = tid & 31;
  const int l16  = lane & 15;     // A: M row / B,C,D: N col
  const int hh   = lane >> 4;     // lane-half: K sub-offset & C/D M-offset

  // ---- warm WGP$ with this wave's weight rows (global_prefetch_b8) ----
  {
    const int ne = wave * 16 + l16;                 // enc N col (phase B)
    const int nd = (2 * wave) * 16 + l16;           // dec N cols (phase C)
    __builtin_prefetch(We + (size_t)ne * DIN, 0, 3);
    __builtin_prefetch(Wd + (size_t)nd * DOUT, 0, 3);
    __builtin_prefetch(Wd + (size_t)(nd + 16) * DOUT, 0, 3);
  }

  // ---- window bounds: union of all 16 nodes' clipped windows ----
  int wlo = s - KHALF;
  if (wlo < 0) wlo = 0;
  if (wlo > N_NODES - 1 - 2 * KHALF) wlo = N_NODES - 1 - 2 * KHALF;

  // ---- stage window rows into LDS ----
#if __has_builtin(__builtin_amdgcn_global_load_async_to_lds_b128)
  for (int idx = tid; idx < WROWS * (DIN / 4); idx += 128) {
    int r  = idx >> 5;           // row in window (DIN/4 == 32 float4 per row)
    int c4 = idx & 31;
    int row = wlo + r; if (row > N_NODES - 1) row = N_NODES - 1;
    const float* gp = h + (size_t)row * DIN + c4 * 4;
    float*       lp = winb + r * WS + c4 * 4;
    __builtin_amdgcn_global_load_async_to_lds_b128((gv4i_p)gp, (lv4i_p)lp, 0, 0);
  }
#if __has_builtin(__builtin_amdgcn_s_wait_asynccnt)
  __builtin_amdgcn_s_wait_asynccnt(0);
#else
  asm volatile("s_wait_asynccnt 0" ::: "memory");
#endif
#else
  for (int idx = tid; idx < WROWS * (DIN / 4); idx += 128) {
    int r  = idx >> 5;
    int c4 = idx & 31;
    int row = wlo + r; if (row > N_NODES - 1) row = N_NODES - 1;
    float4 v = ((const float4*)(h + (size_t)row * DIN))[c4];
    ((float4*)(winb + r * WS))[c4] = v;
  }
#endif
  __syncthreads();

  // ================= Phase A: distances, softmax weights, aggregation =========
  const int nl   = tid >> 3;           // node-local 0..15 (8 lanes per node)
  const int sub  = tid & 7;            // lane's 16-dim slice
  const int nid  = s + nl;

  int left = nid - KHALF;
  if (left < 0) left = 0;
  if (left > N_NODES - 1 - 2 * KHALF) left = N_NODES - 1 - 2 * KHALF;
  const int pos  = nid - left;         // position of nid in its window
  const int rowI = nid - wlo;

  float dist[WINN];
  int   wrow[WINN];
  const float* selfp = winb + rowI * WS + sub * 16;
  #pragma unroll
  for (int j = 0; j < WINN; ++j) {
    int nbr = left + j + (j >= pos ? 1 : 0);   // skip self slot
    int wr  = nbr - wlo;
    wrow[j] = wr;
    const float* np = winb + wr * WS + sub * 16;
    float acc = 0.f;
    #pragma unroll
    for (int d = 0; d < 16; ++d) {
      float df = selfp[d] - np[d];
      acc = fmaf(df, df, acc);
    }
    // reduce across the 8 lanes of this node (aligned subgroup of wave32)
    acc += __shfl_xor(acc, 1, 32);
    acc += __shfl_xor(acc, 2, 32);
    acc += __shfl_xor(acc, 4, 32);
    dist[j] = sqrtf(acc);
  }

  float bsum = 0.f;
  #pragma unroll
  for (int j = 0; j < WINN; ++j) bsum += dist[j];
  const float invb = (float)WINN / bsum;        // 1/beta, beta = mean(dist)

  float mx = -INFINITY;
  float lg[WINN];
  #pragma unroll
  for (int j = 0; j < WINN; ++j) { lg[j] = -dist[j] * invb; mx = fmaxf(mx, lg[j]); }
  float wsum = 0.f, wts[WINN];
  #pragma unroll
  for (int j = 0; j < WINN; ++j) { wts[j] = __expf(lg[j] - mx); wsum += wts[j]; }
  const float winv = 1.f / wsum;
  #pragma unroll
  for (int j = 0; j < WINN; ++j) wts[j] *= winv;

  // agg = h_i + sum_j w_j * h_nbr_j   (each lane: its 16 dims, float4)
  #pragma unroll
  for (int q = 0; q < 4; ++q) {
    int d0 = sub * 16 + q * 4;
    float4 a = *(const float4*)(winb + rowI * WS + d0);
    #pragma unroll
    for (int j = 0; j < WINN; ++j) {
      float4 nv = *(const float4*)(winb + wrow[j] * WS + d0);
      a.x = fmaf(wts[j], nv.x, a.x);
      a.y = fmaf(wts[j], nv.y, a.y);
      a.z = fmaf(wts[j], nv.z, a.z);
      a.w = fmaf(wts[j], nv.w, a.w);
    }
    *(float4*)(aggb + nl * AS + d0) = a;
  }
  __syncthreads();   // after this barrier winb is dead -> reused for dec staging

  // ================= Phase B: enc = relu(agg @ We^T + be), WMMA f32 ===========
  {
    const int n = wave * 16 + l16;            // this wave's N-tile of enc
    const float bias = be[n];
    v8f c = {bias, bias, bias, bias, bias, bias, bias, bias};  // bias via C-init
    #pragma unroll
    for (int k0 = 0; k0 < DIN; k0 += 4) {
      const int k = k0 + 2 * hh;
      v2f a = *(const v2f*)(aggb + l16 * AS + k);          // A[m][k], A[m][k+1]
      v2f b = *(const v2f*)(We + (size_t)n * DIN + k);     // B[k][n] = We[n][k]
      c = wmma_f32(a, b, c);
    }
    #pragma unroll
    for (int r = 0; r < 8; ++r) {
      float v = c[r];
      encb[(r + 8 * hh) * ES + n] = v > 0.f ? v : 0.f;     // ReLU -> LDS
    }
  }
  __syncthreads();

  // ================= Phase C: dec = relu(enc @ Wd^T + bd), WMMA f32 ===========
  const int n0 = (2 * wave) * 16 + l16;       // each wave owns two N-tiles of dec
  const int n1 = (2 * wave + 1) * 16 + l16;
  const float b0v = bd[n0], b1v = bd[n1];
  v8f c0 = {b0v, b0v, b0v, b0v, b0v, b0v, b0v, b0v};
  v8f c1 = {b1v, b1v, b1v, b1v, b1v, b1v, b1v, b1v};
  #pragma unroll
  for (int k0 = 0; k0 < DOUT; k0 += 4) {
    const int k = k0 + 2 * hh;
    v2f a   = *(const v2f*)(encb + l16 * ES + k);          // shared A-fragment
    v2f bb0 = *(const v2f*)(Wd + (size_t)n0 * DOUT + k);   // B[k][n] = Wd[n][k]
    v2f bb1 = *(const v2f*)(Wd + (size_t)n1 * DOUT + k);
    c0 = wmma_f32(a, bb0, c0);
    c1 = wmma_f32(a, bb1, c1);
  }
  // ReLU, stage dec tile into LDS (winb reused; disjoint columns per wave)
  #pragma unroll
  for (int r = 0; r < 8; ++r) {
    const int m = r + 8 * hh;
    float v0 = c0[r]; v0 = v0 > 0.f ? v0 : 0.f;
    float v1 = c1[r]; v1 = v1 > 0.f ? v1 : 0.f;
    winb[m * WS + n0] = v0;
    winb[m * WS + n1] = v1;
  }
  __syncthreads();

  // ---- fully-coalesced float4 writeback of the 16x128 dec tile ----
  #pragma unroll
  for (int it = 0; it < 4; ++it) {
    int idx = tid + it * 128;            // 16 rows * 32 float4 = 512
    int r  = idx >> 5;
    int c4 = idx & 31;
    float4 v = *(const float4*)(winb + r * WS + c4 * 4);
    ((float4*)(out + (size_t)(s + r) * DIN))[c4] = v;
  }
}

extern "C" void kernel_launch(void* const* d_in, const int* in_sizes, int n_in,
                              void* d_out, int out_size, void* d_ws, size_t ws_size,
                              hipStream_t stream) {
  (void)in_sizes; (void)n_in; (void)out_size; (void)d_ws; (void)ws_size;
  const float* h  = (const float*)d_in[0];
  const float* We = (const float*)d_in[1];
  const float* be = (const float*)d_in[2];
  const float* Wd = (const float*)d_in[3];
  const float* bd = (const float*)d_in[4];
  float* out = (float*)d_out;
  const int blocks = N_NODES / TILE;   // 50000 / 16 = 3125, exact
  gae_fused_kernel<<<blocks, 128, 0, stream>>>(h, We, be, Wd, bd, out);
}